// RenderNet_26216480375152
// MI455X (gfx1250) — compile-verified
//
#include <hip/hip_runtime.h>
#include <math.h>

typedef __attribute__((ext_vector_type(2))) float v2f;
typedef __attribute__((ext_vector_type(8))) float v8f;

#define RADIUS   0.1f
#define R2       (RADIUS * RADIUS)
#define INV_R3   (1.0f / (RADIUS * RADIUS * RADIUS))
#define KNN      32
#define MAXPART  4096
#define EPSW     1e-12f

#define QPB      128        // queries per block (8 waves * 16)
#define THREADS  256

// raw v_sqrt_f32 (1 ULP, TRANS pipe)
__device__ __forceinline__ float fast_sqrt(float x) {
    return __builtin_amdgcn_sqrtf(x);
}
// single v_max_num_f32 without the compiler's canonicalize-of-uncanonical-input
__device__ __forceinline__ float max0_asm(float x) {
    float r;
    asm("v_max_num_f32 %0, 0, %1" : "=v"(r) : "v"(x));
    return r;
}
// single v_med3_f32: clamp to [0,1] (upper bound inert for w = 1-(d/R)^3, d>=0)
__device__ __forceinline__ float clamp01(float x) {
    return __builtin_amdgcn_fmed3f(x, 0.0f, 1.0f);
}

__global__ __launch_bounds__(THREADS)
void ball_smooth_wmma(const float* __restrict__ q_in,   // [Q][3]
                      const float* __restrict__ p_in,   // [P][3]
                      float* __restrict__ out,          // [Q][4]
                      int Q, int P)
{
    __shared__ float4 sP[MAXPART];   // (x, y, z, |p|^2)   64 KB
    __shared__ float4 sQ[QPB];       // (x, y, z, |q|^2)    2 KB

    const int tid = threadIdx.x;

    // ---- stage particles into LDS with precomputed |p|^2 ----
    for (int i = tid; i < P; i += THREADS) {
        float x = p_in[3 * i + 0];
        float y = p_in[3 * i + 1];
        float z = p_in[3 * i + 2];
        sP[i] = make_float4(x, y, z, fmaf(x, x, fmaf(y, y, z * z)));
    }
    const int qbase = blockIdx.x * QPB;
    if (tid < QPB) {
        int qi = qbase + tid;
        float x = 0.f, y = 0.f, z = 0.f;
        if (qi < Q) { x = q_in[3 * qi + 0]; y = q_in[3 * qi + 1]; z = q_in[3 * qi + 2]; }
        sQ[tid] = make_float4(x, y, z, fmaf(x, x, fmaf(y, y, z * z)));
    }
    __syncthreads();

    const int lane  = tid & 31;
    const int wave  = tid >> 5;
    const int laneN = lane & 15;        // B/C column (particle), A row (query)
    const bool hi   = lane >= 16;
    const int qoff  = wave * 16;        // this wave's 16 queries within block

    // ---- A operand: -2*q so that WMMA(A,B)+C gives d^2 directly ----
    float4 qa = sQ[qoff + laneN];
    v2f A;
    A.x = hi ? (-2.0f * qa.z) : (-2.0f * qa.x);   // K=2 : K=0
    A.y = hi ? 0.0f           : (-2.0f * qa.y);   // K=3 : K=1

    // |q|^2 for the 8 query rows this lane-half owns in C/D layout
    float q2v[8];
    const int mbase = qoff + (hi ? 8 : 0);
#pragma unroll
    for (int v = 0; v < 8; ++v) q2v[v] = sQ[mbase + v].w;

    // accumulators: per lane, 8 query rows x (w, wx, wy, wz, count-of-positive-w)
    float sw[8], sx[8], sy[8], sz[8], cn[8];
#pragma unroll
    for (int v = 0; v < 8; ++v) { sw[v] = sx[v] = sy[v] = sz[v] = cn[v] = 0.f; }

    const int ntiles = P >> 4;
#pragma unroll 2
    for (int t = 0; t < ntiles; ++t) {
        float4 p = sP[(t << 4) + laneN];          // broadcast within half
        v2f B;
        B.x = hi ? p.z : p.x;                     // K=2 : K=0
        B.y = hi ? 0.f : p.y;                     // K=3 : K=1
        v8f c;
#pragma unroll
        for (int v = 0; v < 8; ++v) c[v] = q2v[v] + p.w;   // C = |q|^2 + |p|^2
        // d2[M][N] = |q_M|^2 + |p_N|^2 - 2 q_M . p_N
        c = __builtin_amdgcn_wmma_f32_16x16x4_f32(false, A, false, B,
                                                  (short)0, c, false, false);
#pragma unroll
        for (int v = 0; v < 8; ++v) {
            float d2 = max0_asm(c[v]);                            // 1x v_max
            float d  = fast_sqrt(d2);                             // v_sqrt (TRANS)
            float w  = clamp01(fmaf(d2 * d, -INV_R3, 1.0f));      // 1x v_med3
            sw[v] += w;
            sx[v]  = fmaf(w, p.x, sx[v]);
            sy[v]  = fmaf(w, p.y, sy[v]);
            sz[v]  = fmaf(w, p.z, sz[v]);
            cn[v] += ceilf(w);                                    // 1 iff w > 0
        }
    }

    // ---- reduce across the 16 lanes of each half (xor masks stay in-half) ----
#pragma unroll
    for (int m = 1; m <= 8; m <<= 1) {
#pragma unroll
        for (int v = 0; v < 8; ++v) {
            sw[v] += __shfl_xor(sw[v], m, 32);
            sx[v] += __shfl_xor(sx[v], m, 32);
            sy[v] += __shfl_xor(sy[v], m, 32);
            sz[v] += __shfl_xor(sz[v], m, 32);
            cn[v] += __shfl_xor(cn[v], m, 32);
        }
    }

    // ---- lane j in {0..7} of each half finalizes query M = 8*hi + j ----
    const int j = lane & 7;
    float fw = sw[0], fx = sx[0], fy = sy[0], fz = sz[0], fc = cn[0];
#pragma unroll
    for (int v = 1; v < 8; ++v)
        if (j == v) { fw = sw[v]; fx = sx[v]; fy = sy[v]; fz = sz[v]; fc = cn[v]; }

    const bool writer = (lane < 8) || (hi && (lane < 24));
    if (writer) {
        const int M  = (hi ? 8 : 0) + j;
        const int qi = qbase + qoff + M;
        if (qi < Q) {
            float4 qd = sQ[qoff + M];
            int n = (int)(fc + 0.5f);

            if (n > KNN) {
                // rare exact path: true 32-smallest selection for this query
                float best[KNN];
#pragma unroll
                for (int k = 0; k < KNN; ++k) best[k] = 3.0e38f;
                float mx = 3.0e38f;
                for (int p = 0; p < P; ++p) {
                    float4 pp = sP[p];
                    float dx = pp.x - qd.x, dy = pp.y - qd.y, dz = pp.z - qd.z;
                    float d2 = fmaf(dx, dx, fmaf(dy, dy, dz * dz));
                    if (d2 <= R2 && d2 < mx) {
                        int am = 0; float cm = best[0];
                        for (int k = 1; k < KNN; ++k) if (best[k] > cm) { cm = best[k]; am = k; }
                        best[am] = d2;
                        mx = best[0];
                        for (int k = 1; k < KNN; ++k) mx = fmaxf(mx, best[k]);
                    }
                }
                const float t2 = mx;   // 32nd-smallest d^2
                fw = fx = fy = fz = 0.f;
                int cnt = 0;
                for (int p = 0; p < P && cnt < KNN; ++p) {
                    float4 pp = sP[p];
                    float dx = pp.x - qd.x, dy = pp.y - qd.y, dz = pp.z - qd.z;
                    float d2 = fmaf(dx, dx, fmaf(dy, dy, dz * dz));
                    if (d2 <= t2) {
                        float d = fast_sqrt(d2);
                        float w = clamp01(fmaf(d2 * d, -INV_R3, 1.0f));
                        fw += w;
                        fx = fmaf(w, pp.x, fx);
                        fy = fmaf(w, pp.y, fy);
                        fz = fmaf(w, pp.z, fz);
                        ++cnt;
                    }
                }
                n = KNN;
            }

            // masked slots: neighbor==0 -> nn_d = |q|, weight w0 each
            const int   m   = KNN - n;
            const float qn  = fast_sqrt(qd.w);
            const float w0  = clamp01(fmaf(qd.w * qn, -INV_R3, 1.0f));
            const float dens = fw + (float)m * w0;
            const float inv  = 1.0f / (dens + EPSW);
            ((float4*)out)[qi] = make_float4(fx * inv, fy * inv, fz * inv, dens);
        }
    }
}

extern "C" void kernel_launch(void* const* d_in, const int* in_sizes, int n_in,
                              void* d_out, int out_size, void* d_ws, size_t ws_size,
                              hipStream_t stream) {
    const float* q = (const float*)d_in[0];   // ray_particles [512*64*3]
    const float* p = (const float*)d_in[1];   // particles    [4096*3]
    float* out = (float*)d_out;               // [512*64*4]
    const int Q = in_sizes[0] / 3;            // 32768
    const int P = in_sizes[1] / 3;            // 4096
    const int blocks = (Q + QPB - 1) / QPB;   // 256
    ball_smooth_wmma<<<blocks, THREADS, 0, stream>>>(q, p, out, Q, P);
}